// SGCN_33088428048395
// MI455X (gfx1250) — compile-verified
//
#include <hip/hip_runtime.h>
#include <hip/hip_bf16.h>

// SGConv k=1 on MI455X (gfx1250).
// out[n] = norm[n] * sum_{e: dst[e]=n} norm[src[e]] * (feats[src[e]] @ W^T) + b
// Linear layer commutes with aggregation -> project 64->40 BEFORE the edge
// phase (1.6x less gather/scatter traffic; Z/agg are L2-resident on 192MB L2).
// GEMM uses full-precision V_WMMA_F32_16X16X4_F32 (f32 in, f32 acc).

#define IN_DIM 64
#define OUT_DIM 40

typedef __attribute__((ext_vector_type(2))) float v2f;
typedef __attribute__((ext_vector_type(8))) float v8f;

// ---- zero d_out (used as agg accumulator) and deg ----
__global__ void sgc_zero(float* __restrict__ out, float* __restrict__ deg,
                         int out_total, int n_nodes) {
  int t = blockIdx.x * blockDim.x + threadIdx.x;
  if (t < out_total) out[t] = 0.0f;
  if (t < n_nodes)   deg[t] = 0.0f;
}

// ---- deg[dst[e]] += 1 ----
__global__ void sgc_degree(const int* __restrict__ dst, float* __restrict__ deg,
                           int n_edges) {
  int t = blockIdx.x * blockDim.x + threadIdx.x;
  if (t < n_edges) atomicAdd(&deg[dst[t]], 1.0f);
}

// ---- deg -> norm = 1/sqrt(max(deg,1)) in place (full precision) ----
__global__ void sgc_norm(float* __restrict__ deg, int n_nodes) {
  int t = blockIdx.x * blockDim.x + threadIdx.x;
  if (t < n_nodes) deg[t] = 1.0f / sqrtf(fmaxf(deg[t], 1.0f));
}

// ---- Z[n][c] = norm[n] * sum_k feats[n][k] * W[c][k], via WMMA f32 16x16x4.
// One wave per 16-node tile; 3 column tiles (cols 0..47, masked to 40);
// K=64 in 16 steps of 4. W staged transposed in LDS: lw[k*40+n] = W[n][k].
__global__ void __launch_bounds__(256)
sgc_gemm_wmma(const float* __restrict__ feats, const float* __restrict__ W,
              const float* __restrict__ norm, float* __restrict__ Z,
              int n_nodes) {
  __shared__ float lw[IN_DIM * OUT_DIM];
  for (int i = threadIdx.x; i < IN_DIM * OUT_DIM; i += blockDim.x) {
    int k = i / OUT_DIM;
    int n = i - k * OUT_DIM;
    lw[i] = W[n * IN_DIM + k];
  }
  __syncthreads();

  const int lane   = threadIdx.x & 31;
  const int wave   = threadIdx.x >> 5;
  const int ntiles = (n_nodes + 15) >> 4;
  const int tile   = blockIdx.x * (blockDim.x >> 5) + wave;
  if (tile >= ntiles) return;
  const int m0 = tile << 4;

  // A fragment (16x4 f32): lanes 0-15 hold K={0,1}, lanes 16-31 hold K={2,3},
  // row M = lane&15 (one VGPR per K within the pair).
  const int half = lane >> 4;            // 0 -> K pair (0,1); 1 -> (2,3)
  const int mrow = lane & 15;
  int mr = m0 + mrow;
  if (mr >= n_nodes) mr = n_nodes - 1;   // clamp (N%16==0 in practice)
  const float* arow = feats + (size_t)mr * IN_DIM;

  // B fragment (4x16 f32): row K striped across lanes within a VGPR;
  // N = lane&15 within each column tile. Tile 2 masked beyond col 39.
  const int   ncol  = lane & 15;
  const int   n2    = 32 + ncol;
  const float cmask = (n2 < OUT_DIM) ? 1.0f : 0.0f;
  const int   n2c   = (n2 < OUT_DIM) ? n2 : (OUT_DIM - 1);

  v8f acc0 = {}, acc1 = {}, acc2 = {};
#pragma unroll
  for (int kk = 0; kk < IN_DIM; kk += 4) {
    const int kb = kk + 2 * half;
    v2f a;
    a.x = arow[kb];
    a.y = arow[kb + 1];

    v2f b0, b1, b2;
    b0.x = lw[kb * OUT_DIM + ncol];
    b0.y = lw[(kb + 1) * OUT_DIM + ncol];
    b1.x = lw[kb * OUT_DIM + 16 + ncol];
    b1.y = lw[(kb + 1) * OUT_DIM + 16 + ncol];
    b2.x = lw[kb * OUT_DIM + n2c] * cmask;
    b2.y = lw[(kb + 1) * OUT_DIM + n2c] * cmask;

    acc0 = __builtin_amdgcn_wmma_f32_16x16x4_f32(false, a, false, b0,
                                                 (short)0, acc0, false, false);
    acc1 = __builtin_amdgcn_wmma_f32_16x16x4_f32(false, a, false, b1,
                                                 (short)0, acc1, false, false);
    acc2 = __builtin_amdgcn_wmma_f32_16x16x4_f32(false, a, false, b2,
                                                 (short)0, acc2, false, false);
  }

  // D layout (16x16 f32): VGPR v, lane l -> (M = v + 8*(l>=16), N = l&15).
  // Fuse the src-side norm scale into the store.
#pragma unroll
  for (int v = 0; v < 8; ++v) {
    int m = m0 + v + 8 * half;
    if (m < n_nodes) {
      float  s  = norm[m];
      float* zr = Z + (size_t)m * OUT_DIM;
      zr[ncol]      = acc0[v] * s;
      zr[16 + ncol] = acc1[v] * s;
      if (n2 < OUT_DIM) zr[n2] = acc2[v] * s;
    }
  }
}

// ---- agg[dst[e]][d] += Z[src[e]][d]; one thread per (edge, dim) ----
__global__ void sgc_scatter(const float* __restrict__ Z,
                            const int* __restrict__ src,
                            const int* __restrict__ dst,
                            float* __restrict__ agg, int n_edges) {
  long long t     = (long long)blockIdx.x * blockDim.x + threadIdx.x;
  long long total = (long long)n_edges * OUT_DIM;
  if (t >= total) return;
  int e = (int)(t / OUT_DIM);
  int d = (int)(t - (long long)e * OUT_DIM);
  int s = src[e];
  int q = dst[e];
  atomicAdd(&agg[(size_t)q * OUT_DIM + d], Z[(size_t)s * OUT_DIM + d]);
}

// ---- out = out * norm[n] + b[d], in place ----
__global__ void sgc_finalize(float* __restrict__ out,
                             const float* __restrict__ norm,
                             const float* __restrict__ bias, int n_nodes) {
  int t     = blockIdx.x * blockDim.x + threadIdx.x;
  int total = n_nodes * OUT_DIM;
  if (t >= total) return;
  int n = t / OUT_DIM;
  int d = t - n * OUT_DIM;
  out[t] = out[t] * norm[n] + bias[d];
}

extern "C" void kernel_launch(void* const* d_in, const int* in_sizes, int n_in,
                              void* d_out, int out_size, void* d_ws,
                              size_t ws_size, hipStream_t stream) {
  const float* feats = (const float*)d_in[0];
  const float* W     = (const float*)d_in[1];
  const float* bias  = (const float*)d_in[2];
  const int*   src   = (const int*)d_in[3];
  const int*   dst   = (const int*)d_in[4];

  const int n_nodes = in_sizes[0] / IN_DIM;
  const int n_edges = in_sizes[3];
  float*    out     = (float*)d_out;

  // Workspace: [deg/norm : n_nodes floats][Z : n_nodes*40 floats]
  float* deg = (float*)d_ws;
  float* Z   = deg + (size_t)((n_nodes + 255) & ~255);

  const int out_total = n_nodes * OUT_DIM;

  sgc_zero<<<(out_total + 255) / 256, 256, 0, stream>>>(out, deg, out_total,
                                                        n_nodes);
  sgc_degree<<<(n_edges + 255) / 256, 256, 0, stream>>>(dst, deg, n_edges);
  sgc_norm<<<(n_nodes + 255) / 256, 256, 0, stream>>>(deg, n_nodes);

  {
    int ntiles = (n_nodes + 15) / 16;
    int wpb    = 256 / 32;  // 8 waves (tiles) per block
    int blocks = (ntiles + wpb - 1) / wpb;
    sgc_gemm_wmma<<<blocks, 256, 0, stream>>>(feats, W, deg, Z, n_nodes);
  }
  {
    long long total  = (long long)n_edges * OUT_DIM;
    int       blocks = (int)((total + 255) / 256);
    sgc_scatter<<<blocks, 256, 0, stream>>>(Z, src, dst, out, n_edges);
  }
  sgc_finalize<<<(out_total + 255) / 256, 256, 0, stream>>>(out, deg, bias,
                                                            n_nodes);
}